// SimOTAAssigner_13460427505716
// MI455X (gfx1250) — compile-verified
//
#include <hip/hip_runtime.h>
#include <hip/hip_bf16.h>
#include <stdint.h>

// Problem constants (match reference)
#define NB   16
#define NM   128
#define NA   8400
#define NCLS 80
#define NBA  (NB * NA)
#define TK   10
#define BIGC 100000.0f
#define INFC 1000000000.0f

// ---- CDNA5 async global->LDS path (guarded; falls back to direct loads) ----
#if defined(__gfx1250__) && \
    __has_builtin(__builtin_amdgcn_global_load_async_to_lds_b128) && \
    __has_builtin(__builtin_amdgcn_global_load_async_to_lds_b64)  && \
    __has_builtin(__builtin_amdgcn_s_wait_asynccnt)
#define USE_ASYNC_LDS 1
typedef int v4i __attribute__((vector_size(16)));
typedef int v2i __attribute__((vector_size(8)));
#define GAS __attribute__((address_space(1)))
#define LAS __attribute__((address_space(3)))
#else
#define USE_ASYNC_LDS 0
#endif

__device__ __forceinline__ float iou_fn(float px1, float py1, float px2, float py2,
                                        float gx1, float gy1, float gx2, float gy2) {
  float lx = fmaxf(px1, gx1), ly = fmaxf(py1, gy1);
  float rx = fminf(px2, gx2), ry = fminf(py2, gy2);
  float w = fmaxf(rx - lx, 0.f), h = fmaxf(ry - ly, 0.f);
  float inter = w * h;
  float a1 = (px2 - px1) * (py2 - py1);
  float a2 = (gx2 - gx1) * (gy2 - gy1);
  return inter / (a1 + a2 - inter + 1e-9f);
}

// One block per (batch, gt). 256 threads = 8 wave32s.
// Double-buffered async global->LDS staging of (pd_bbox, anchor) tiles,
// pipelined with s_wait_asynccnt(2); score-column gather prefetched one
// chunk ahead via global_prefetch.
__global__ __launch_bounds__(256) void k_assign(
    const float* __restrict__ scores, const float* __restrict__ pboxes,
    const float* __restrict__ anc,    const int*   __restrict__ glab,
    const float* __restrict__ gbox,   const float* __restrict__ gmask,
    unsigned long long* __restrict__ best)
{
  const int blk = blockIdx.x;
  const int b = blk >> 7;     // / NM (==128)
  const int m = blk & 127;    // % NM
  if (gmask[b * NM + m] <= 0.f) return;  // invalid GT contributes no candidates
  const int t = threadIdx.x;

  const float gx1 = gbox[(b * NM + m) * 4 + 0];
  const float gy1 = gbox[(b * NM + m) * 4 + 1];
  const float gx2 = gbox[(b * NM + m) * 4 + 2];
  const float gy2 = gbox[(b * NM + m) * 4 + 3];
  const int   lab = glab[b * NM + m];
  const float* sb = scores + ((size_t)b * NA) * NCLS + lab;  // gathered column
  const float* pb = pboxes + ((size_t)b * NA) * 4;

  // Per-thread sorted top-10 lists (static indexing only -> stays in VGPRs).
  unsigned long long ck[TK];  // ascending cost keys (costbits<<32 | anchor)
  float iv[TK];               // descending IoUs
#pragma unroll
  for (int i = 0; i < TK; ++i) { ck[i] = ~0ull; iv[i] = -1.f; }

  __shared__ unsigned long long sk[256 * TK];  // 20 KB
  __shared__ float              si[256 * TK];  // 10 KB
#if USE_ASYNC_LDS
  __shared__ float pdT[2][256 * 4];            // double-buffered bbox tiles
  __shared__ float anT[2][256 * 2];            // double-buffered anchor tiles
#endif

  const int NCH = (NA + 255) / 256;  // 33 chunks

#if USE_ASYNC_LDS
  {  // pre-issue chunk 0 (t < 256 <= NA, no clamp needed)
    __builtin_amdgcn_global_load_async_to_lds_b128(
        (GAS v4i*)(uintptr_t)(pb + (size_t)t * 4),
        (LAS v4i*)(uint32_t)(uintptr_t)(&pdT[0][t * 4]), 0, 0);
    __builtin_amdgcn_global_load_async_to_lds_b64(
        (GAS v2i*)(uintptr_t)(anc + (size_t)t * 2),
        (LAS v2i*)(uint32_t)(uintptr_t)(&anT[0][t * 2]), 0, 0);
  }
#endif

  for (int c = 0; c < NCH; ++c) {
    const int a = c * 256 + t;
#if USE_ASYNC_LDS
    const int cur = c & 1;
    if (c + 1 < NCH) {
      // Unconditional clamped issue keeps exactly 2 async ops per chunk per
      // wave, so the in-order ASYNCcnt <= 2 wait below is exact.
      int an = (c + 1) * 256 + t;
      an = an < NA ? an : NA - 1;
      const int nxt = cur ^ 1;
      __builtin_amdgcn_global_load_async_to_lds_b128(
          (GAS v4i*)(uintptr_t)(pb + (size_t)an * 4),
          (LAS v4i*)(uint32_t)(uintptr_t)(&pdT[nxt][t * 4]), 0, 0);
      __builtin_amdgcn_global_load_async_to_lds_b64(
          (GAS v2i*)(uintptr_t)(anc + (size_t)an * 2),
          (LAS v2i*)(uint32_t)(uintptr_t)(&anT[nxt][t * 2]), 0, 0);
      __builtin_amdgcn_s_wait_asynccnt(2);  // drain chunk c, keep c+1 in flight
    } else {
      __builtin_amdgcn_s_wait_asynccnt(0);
    }
#endif
    if (a < NA) {
      // Prefetch next chunk's gathered score element (stride NCLS*4 B).
      {
        int ap = a + 256;
        ap = ap < NA ? ap : a;
        __builtin_prefetch((const void*)(sb + (size_t)ap * NCLS), 0, 0);
      }
      float px1, py1, px2, py2, ax, ay;
#if USE_ASYNC_LDS
      px1 = pdT[cur][t * 4 + 0]; py1 = pdT[cur][t * 4 + 1];
      px2 = pdT[cur][t * 4 + 2]; py2 = pdT[cur][t * 4 + 3];
      ax  = anT[cur][t * 2 + 0]; ay  = anT[cur][t * 2 + 1];
#else
      px1 = pb[(size_t)a * 4 + 0]; py1 = pb[(size_t)a * 4 + 1];
      px2 = pb[(size_t)a * 4 + 2]; py2 = pb[(size_t)a * 4 + 3];
      ax  = anc[(size_t)a * 2 + 0]; ay = anc[(size_t)a * 2 + 1];
#endif
      const float iou = iou_fn(px1, py1, px2, py2, gx1, gy1, gx2, gy2);
      const float s  = sb[(size_t)a * NCLS];
      const float x  = -s;                                      // softplus(-s)
      const float sp = fmaxf(x, 0.f) + log1pf(expf(-fabsf(x)));
      const float dmin = fminf(fminf(ax - gx1, ay - gy1), fminf(gx2 - ax, gy2 - ay));
      const float cost = sp + 3.0f * (1.0f - iou) + ((dmin > 1e-9f) ? 0.f : BIGC);

      // Branch-free sorted insert into top-10 smallest keys.
      unsigned long long key = (((unsigned long long)__float_as_uint(cost)) << 32) | (unsigned)a;
#pragma unroll
      for (int i = 0; i < TK; ++i) {
        const bool lt = key < ck[i];
        const unsigned long long lo = lt ? key : ck[i];
        const unsigned long long hi = lt ? ck[i] : key;
        ck[i] = lo; key = hi;
      }
      // Branch-free sorted insert into top-10 largest IoUs.
      float v = iou;
#pragma unroll
      for (int i = 0; i < TK; ++i) {
        const bool gt = v > iv[i];
        const float hi = gt ? v : iv[i];
        const float lo = gt ? iv[i] : v;
        iv[i] = hi; v = lo;
      }
    }
  }

#pragma unroll
  for (int i = 0; i < TK; ++i) { sk[t * TK + i] = ck[i]; si[t * TK + i] = iv[i]; }
  __syncthreads();

  // Tree merge of 256 sorted-10 lists -> exact global top-10s at slot 0.
  for (int off = 128; off > 0; off >>= 1) {
    if (t < off) {
      unsigned long long aL[TK], bL[TK], rL[TK];
      float aF[TK], bF[TK], rF[TK];
#pragma unroll
      for (int i = 0; i < TK; ++i) {
        aL[i] = sk[t * TK + i]; bL[i] = sk[(t + off) * TK + i];
        aF[i] = si[t * TK + i]; bF[i] = si[(t + off) * TK + i];
      }
      int i0 = 0, j0 = 0, i1 = 0, j1 = 0;
#pragma unroll
      for (int n = 0; n < TK; ++n) {
        const unsigned long long av = aL[i0], bv = bL[j0];
        if (av <= bv) { rL[n] = av; ++i0; } else { rL[n] = bv; ++j0; }
        const float af = aF[i1], bf = bF[j1];
        if (af >= bf) { rF[n] = af; ++i1; } else { rF[n] = bf; ++j1; }
      }
#pragma unroll
      for (int i = 0; i < TK; ++i) { sk[t * TK + i] = rL[i]; si[t * TK + i] = rF[i]; }
    }
    __syncthreads();
  }

  if (t == 0) {
    float ssum = 0.f;
#pragma unroll
    for (int i = 0; i < TK; ++i) ssum += si[i];     // sum of top-10 IoUs
    int k = (int)floorf(ssum);
    k = k < 1 ? 1 : (k > TK ? TK : k);
    for (int i = 0; i < k; ++i) {
      const unsigned long long key = sk[i];
      const unsigned a = (unsigned)(key & 0xffffffffu);
      const unsigned long long nk = (key & 0xffffffff00000000ull) | (unsigned long long)(unsigned)m;
      atomicMin(best + (size_t)b * NA + a, nk);      // min over (cost, then m)
    }
  }
}

__global__ void k_init(unsigned long long* __restrict__ best) {
  int i = blockIdx.x * blockDim.x + threadIdx.x;
  if (i < NBA) best[i] = ((unsigned long long)__float_as_uint(INFC)) << 32; // (INF, m=0)
}

__global__ void k_finalize(
    const unsigned long long* __restrict__ best,
    const float* __restrict__ pboxes, const int* __restrict__ glab,
    const float* __restrict__ gbox,
    float* __restrict__ o_lab, float* __restrict__ o_box,
    float* __restrict__ o_fg,  float* __restrict__ o_tg,
    int2* __restrict__ meta)
{
  const int idx = blockIdx.x * blockDim.x + threadIdx.x;
  if (idx >= NBA) return;
  const int b = idx / NA;
  const unsigned long long key = best[idx];
  const float cost = __uint_as_float((unsigned)(key >> 32));
  const int m = (int)(key & 0xffffffffu);
  const bool matched = cost < INFC;

  float lab_out = (float)NCLS, fg = 0.f, tg = 0.f;
  float bx = 0.f, by = 0.f, bz = 0.f, bw = 0.f;
  int labm = -1; float val = 0.f;
  if (matched) {
    const float gx1 = gbox[(b * NM + m) * 4 + 0];
    const float gy1 = gbox[(b * NM + m) * 4 + 1];
    const float gx2 = gbox[(b * NM + m) * 4 + 2];
    const float gy2 = gbox[(b * NM + m) * 4 + 3];
    labm = glab[b * NM + m];
    const float px1 = pboxes[(size_t)idx * 4 + 0];
    const float py1 = pboxes[(size_t)idx * 4 + 1];
    const float px2 = pboxes[(size_t)idx * 4 + 2];
    const float py2 = pboxes[(size_t)idx * 4 + 3];
    val = iou_fn(px1, py1, px2, py2, gx1, gy1, gx2, gy2);  // iou_t[best_gt, a]
    lab_out = (float)labm; fg = 1.f; tg = (float)m;
    bx = gx1; by = gy1; bz = gx2; bw = gy2;
  }
  o_lab[idx] = lab_out;
  float4 bb; bb.x = bx; bb.y = by; bb.z = bz; bb.w = bw;
  ((float4*)o_box)[idx] = bb;
  o_fg[idx] = fg;
  o_tg[idx] = tg;
  meta[idx] = make_int2(labm, __float_as_int(val));
}

__global__ void k_scores(const int2* __restrict__ meta, float* __restrict__ o_sc) {
  const int QP = NCLS / 4;  // 20 float4s per anchor row
  const int tid = blockIdx.x * blockDim.x + threadIdx.x;
  if (tid >= NBA * QP) return;
  const int anchor = tid / QP;
  const int q = tid - anchor * QP;
  const int2 mv = meta[anchor];
  const int labm = mv.x;
  const float val = __int_as_float(mv.y);
  const int c0 = q * 4;
  float4 o;
  o.x = (c0 + 0 == labm) ? val : 0.f;
  o.y = (c0 + 1 == labm) ? val : 0.f;
  o.z = (c0 + 2 == labm) ? val : 0.f;
  o.w = (c0 + 3 == labm) ? val : 0.f;
  ((float4*)(o_sc + (size_t)anchor * NCLS))[q] = o;
}

extern "C" void kernel_launch(void* const* d_in, const int* in_sizes, int n_in,
                              void* d_out, int out_size, void* d_ws, size_t ws_size,
                              hipStream_t stream) {
  (void)in_sizes; (void)n_in; (void)out_size; (void)ws_size;
  const float* pd_scores = (const float*)d_in[0];  // (B, A, NC)
  const float* pd_bboxes = (const float*)d_in[1];  // (B, A, 4)
  const float* anc       = (const float*)d_in[2];  // (A, 2)
  const int*   gt_labels = (const int*)d_in[3];    // (B, M, 1)
  const float* gt_bboxes = (const float*)d_in[4];  // (B, M, 4)
  const float* mask_gt   = (const float*)d_in[5];  // (B, M, 1)

  const size_t BA = (size_t)NBA;
  float* out   = (float*)d_out;
  float* o_lab = out;                       // B*A
  float* o_box = out + BA;                  // B*A*4
  float* o_sc  = out + BA * 5;              // B*A*NC
  float* o_fg  = out + BA * (5 + NCLS);     // B*A
  float* o_tg  = out + BA * (6 + NCLS);     // B*A

  unsigned long long* best = (unsigned long long*)d_ws;           // 8B * B*A
  int2* meta = (int2*)((char*)d_ws + BA * sizeof(unsigned long long));

  k_init<<<(NBA + 255) / 256, 256, 0, stream>>>(best);
  k_assign<<<NB * NM, 256, 0, stream>>>(pd_scores, pd_bboxes, anc,
                                        gt_labels, gt_bboxes, mask_gt, best);
  k_finalize<<<(NBA + 255) / 256, 256, 0, stream>>>(best, pd_bboxes, gt_labels,
                                                    gt_bboxes, o_lab, o_box, o_fg,
                                                    o_tg, meta);
  k_scores<<<((NBA * (NCLS / 4)) + 255) / 256, 256, 0, stream>>>(meta, o_sc);
}